// FocalLoss_64759516889772
// MI455X (gfx1250) — compile-verified
//
#include <hip/hip_runtime.h>

// ---------------------------------------------------------------------------
// Focal loss, N=1048576 rows x C=128 classes, fp32 preds + int32 one-hot.
// Memory-bound: 1 GB streamed once -> ~44us floor at 23.3 TB/s.
// One wave32 per row: each lane loads one float4 of preds + one int4 of
// labels (fully coalesced 512B/row), exp via v_exp_f32, wave-wide sums via
// V_WMMA_F32_16X16X4_F32 (all-ones B => D[m][n] = sum_k A[m][k]).
// Loop unrolled x2 over independent rows for MLP; division eliminated via
// p = exp(-(log t1 - lbl)). Deterministic two-pass mean through d_ws.
// ---------------------------------------------------------------------------

typedef float f32x4 __attribute__((ext_vector_type(4)));
typedef int   i32x4 __attribute__((ext_vector_type(4)));
typedef float v2f   __attribute__((ext_vector_type(2)));
typedef float v8f   __attribute__((ext_vector_type(8)));

#define WAVES_PER_BLOCK 8
#define BLOCK_THREADS   256
#define GRID_BLOCKS     4096
#define NCLASS          128
#define ALPHA_C         0.25f

// Wave-wide (32 lane) float sum on the f32 WMMA unit.
// A vgpr0 = lane value, vgpr1 = 0; B = all-ones. Every D element is a row-sum
// of A; D's documented layout puts rows 0-7 in lanes 0-15 and rows 8-15 in
// lanes 16-31, so sum(8 D regs) + shfl_xor(16) = sum over all 32 lane values
// in every lane, independent of A's internal (m,k) striping.
__device__ __forceinline__ float wave_reduce_sum32(float v) {
  v2f a; a[0] = v;    a[1] = 0.0f;
  v2f b; b[0] = 1.0f; b[1] = 1.0f;
  v8f c = {};
  c = __builtin_amdgcn_wmma_f32_16x16x4_f32(
      /*neg_a=*/false, a, /*neg_b=*/false, b,
      /*c_mod=*/(short)0, c, /*reuse_a=*/false, /*reuse_b=*/false);
  float s = c[0] + c[1] + c[2] + c[3] + c[4] + c[5] + c[6] + c[7];
  s += __shfl_xor(s, 16, 32);
  return s;
}

// Per-row focal loss: u = log(sum_j exp) - label_logit, p = exp(-u),
// loss = alpha * (1-p)^2 * u.   (identical to alpha*(1-e^l/t1)^2*(log t1 - l))
__device__ __forceinline__ float row_loss(const float* __restrict__ preds,
                                          const int*   __restrict__ labels,
                                          int row, int lane) {
  const f32x4* prow = (const f32x4*)(preds  + (size_t)row * NCLASS);
  const i32x4* lrow = (const i32x4*)(labels + (size_t)row * NCLASS);
  // Streaming data (1 GB >> 192 MB L2): non-temporal b128 loads.
  f32x4 p = __builtin_nontemporal_load(prow + lane);
  i32x4 l = __builtin_nontemporal_load(lrow + lane);

  float e = __expf(p.x) + __expf(p.y) + __expf(p.z) + __expf(p.w);
  float g = (l.x ? p.x : 0.0f) + (l.y ? p.y : 0.0f) +
            (l.z ? p.z : 0.0f) + (l.w ? p.w : 0.0f);

  float t1  = wave_reduce_sum32(e);   // sum_j exp(preds[row, j])
  float lbl = wave_reduce_sum32(g);   // preds[row, label_idx]

  float u  = __logf(t1) - lbl;        // = -log p
  float om = 1.0f - __expf(-u);       // = 1 - p  (no divide)
  return ALPHA_C * om * om * u;
}

__global__ void focal_partial_kernel(const float* __restrict__ preds,
                                     const int*   __restrict__ labels,
                                     float*       __restrict__ block_part,
                                     int nrows) {
  __shared__ float wsum[WAVES_PER_BLOCK];
  const int lane   = threadIdx.x & 31;
  const int wave   = threadIdx.x >> 5;
  const int gwave  = blockIdx.x * WAVES_PER_BLOCK + wave;
  const int nwaves = gridDim.x * WAVES_PER_BLOCK;

  // Row index is wave-uniform -> EXEC is all-ones inside the loop (WMMA req).
  float acc0 = 0.0f, acc1 = 0.0f;
  int row = gwave;
  for (; row + nwaves < nrows; row += 2 * nwaves) {   // 2 independent rows
    acc0 += row_loss(preds, labels, row,          lane);
    acc1 += row_loss(preds, labels, row + nwaves, lane);
  }
  if (row < nrows)
    acc0 += row_loss(preds, labels, row, lane);
  float acc = acc0 + acc1;

  if (lane == 0) wsum[wave] = acc;    // acc is wave-uniform after shfl_xor
  __syncthreads();
  if (threadIdx.x == 0) {
    float s = 0.0f;
#pragma unroll
    for (int i = 0; i < WAVES_PER_BLOCK; ++i) s += wsum[i];
    block_part[blockIdx.x] = s;
  }
}

__global__ void focal_finalize_kernel(const float* __restrict__ parts,
                                      int nparts, float inv_n,
                                      float* __restrict__ out) {
  __shared__ float sm[BLOCK_THREADS];
  float s = 0.0f;
  for (int i = threadIdx.x; i < nparts; i += BLOCK_THREADS) s += parts[i];
  sm[threadIdx.x] = s;
  __syncthreads();
  for (int off = BLOCK_THREADS / 2; off > 0; off >>= 1) {
    if ((int)threadIdx.x < off) sm[threadIdx.x] += sm[threadIdx.x + off];
    __syncthreads();
  }
  if (threadIdx.x == 0) out[0] = sm[0] * inv_n;
}

extern "C" void kernel_launch(void* const* d_in, const int* in_sizes, int n_in,
                              void* d_out, int out_size, void* d_ws, size_t ws_size,
                              hipStream_t stream) {
  (void)n_in; (void)out_size; (void)ws_size;
  const float* preds  = (const float*)d_in[0];
  const int*   labels = (const int*)d_in[1];
  float*       out    = (float*)d_out;
  float*       parts  = (float*)d_ws;          // GRID_BLOCKS floats = 16 KB

  const int nrows = in_sizes[0] / NCLASS;      // 1048576
  const float inv_n = 1.0f / (float)nrows;

  focal_partial_kernel<<<GRID_BLOCKS, BLOCK_THREADS, 0, stream>>>(
      preds, labels, parts, nrows);
  focal_finalize_kernel<<<1, BLOCK_THREADS, 0, stream>>>(
      parts, GRID_BLOCKS, inv_n, out);
}